// DCT2_80599356277181
// MI455X (gfx1250) — compile-verified
//
#include <hip/hip_runtime.h>

// 2-D DCT-II of a 4096x4096 fp32 matrix as C = A * X * A^T with
// A[k][n] = cos(pi*k*(2n+1)/8192), via fp32 WMMA (v_wmma_f32_16x16x4_f32).
//
//   1) dct_basis_kernel : build A into d_ws            (64 MB)
//   2) gemm_nt_wmma     : T = A * X^T  -> d_ws + 64MB  (T[v][m] = sum_n A[v,n] x[m,n])
//   3) gemm_nt_wmma     : C = A * T^T  -> d_out        (C[u][v] = sum_m A[u,m] T[v,m])
//
// Staging uses the CDNA5 async global->LDS path (global_load_async_to_lds_b128,
// ASYNCcnt) with LDS double buffering.

typedef __attribute__((ext_vector_type(2))) float v2f;
typedef __attribute__((ext_vector_type(8))) float v8f;
typedef int v4i __attribute__((vector_size(16)));   // matches builtin param pointee

#define MAT_N 4096
#define BLK   128     // output block per workgroup
#define BK    16      // K-chunk per LDS buffer
#define LPAD  20      // LDS row stride (floats): 16B-aligned rows, conflict-free frags

#if __has_builtin(__builtin_amdgcn_global_load_async_to_lds_b128) && \
    __has_builtin(__builtin_amdgcn_s_wait_asynccnt)
#define USE_ASYNC 1
#else
#define USE_ASYNC 0
#endif

__device__ __forceinline__ void copy16(const float* g, float* l) {
#if USE_ASYNC
    __builtin_amdgcn_global_load_async_to_lds_b128(
        (__attribute__((address_space(1))) v4i*)g,
        (__attribute__((address_space(3))) v4i*)l,
        0, 0);
#else
    *(float4*)l = *(const float4*)g;
#endif
}
__device__ __forceinline__ void wait_async_le4() {
#if USE_ASYNC
    __builtin_amdgcn_s_wait_asynccnt(4);   // 4 newest (next-chunk) may remain in flight
#endif
}
__device__ __forceinline__ void wait_async_0() {
#if USE_ASYNC
    __builtin_amdgcn_s_wait_asynccnt(0);
#endif
}

__global__ __launch_bounds__(256) void dct_basis_kernel(float* __restrict__ A) {
    const unsigned idx = blockIdx.x * 256u + threadIdx.x;
    const unsigned k = idx >> 12;
    const unsigned n = idx & 4095u;
    const unsigned phase = (k * (2u * n + 1u)) & 16383u;  // exact mod-2pi reduction
    A[idx] = cosf((float)phase * 3.83495196971410526e-4f);  // * pi/8192
}

// D = P * Q^T  (all 4096x4096 row-major fp32)
__global__ __launch_bounds__(256) void gemm_nt_wmma(const float* __restrict__ P,
                                                    const float* __restrict__ Q,
                                                    float* __restrict__ D) {
    __shared__ float Pl[2][BLK][LPAD];
    __shared__ float Ql[2][BLK][LPAD];

    const int tid  = threadIdx.x;
    const int lane = tid & 31;
    const int wave = tid >> 5;
    const int lm   = lane & 15;
    const int lh   = lane >> 4;

    const int rowBlock = blockIdx.y * BLK;
    const int colBlock = blockIdx.x * BLK;

    // 8 waves arranged 2x4; each wave owns a 64x32 patch = 4x2 WMMA tiles
    const int wRow = (wave >> 2) * 64;
    const int wCol = (wave & 3) * 32;

    v8f acc[4][2];
#pragma unroll
    for (int r = 0; r < 4; ++r)
#pragma unroll
        for (int c = 0; c < 2; ++c)
            acc[r][c] = (v8f){0.f, 0.f, 0.f, 0.f, 0.f, 0.f, 0.f, 0.f};

    // staging: 128x16 floats/operand/chunk; 256 threads x 2 float4 each per operand
    const int cg = (tid & 3) * 4;   // K-column group 0,4,8,12
    const int r0 = tid >> 2;        // row 0..63 (+64 on second pass)

#define STAGE(buf, kc)                                                          \
    do {                                                                        \
        _Pragma("unroll")                                                       \
        for (int i = 0; i < 2; ++i) {                                           \
            const int r = r0 + i * 64;                                          \
            copy16(&P[(size_t)(rowBlock + r) * MAT_N + (kc) + cg],              \
                   &Pl[buf][r][cg]);                                            \
            copy16(&Q[(size_t)(colBlock + r) * MAT_N + (kc) + cg],              \
                   &Ql[buf][r][cg]);                                            \
        }                                                                       \
    } while (0)

    STAGE(0, 0);

    const int NCHUNK = MAT_N / BK;
    for (int c = 0; c < NCHUNK; ++c) {
        const int buf = c & 1;
        if (c + 1 < NCHUNK) {
            STAGE(buf ^ 1, (c + 1) * BK);   // overlap next chunk with this compute
            wait_async_le4();               // current chunk's 4 copies complete
        } else {
            wait_async_0();
        }
        __syncthreads();

#pragma unroll
        for (int kk = 0; kk < BK; kk += 4) {
            // A frag 16x4: lane L -> P[tile+L%16][kk + 2*(L/16) + t]
            // B frag 4x16: lane L -> Q[tile+L%16][kk + 2*(L/16) + t]
            v2f af[4], bf[2];
#pragma unroll
            for (int r = 0; r < 4; ++r)
                af[r] = *(const v2f*)&Pl[buf][wRow + r * 16 + lm][kk + 2 * lh];
#pragma unroll
            for (int cc = 0; cc < 2; ++cc)
                bf[cc] = *(const v2f*)&Ql[buf][wCol + cc * 16 + lm][kk + 2 * lh];
#pragma unroll
            for (int r = 0; r < 4; ++r)
#pragma unroll
                for (int cc = 0; cc < 2; ++cc)
                    acc[r][cc] = __builtin_amdgcn_wmma_f32_16x16x4_f32(
                        false, af[r], false, bf[cc], (short)0, acc[r][cc],
                        false, false);
        }
        __syncthreads();
    }
#undef STAGE

    // C/D layout: VGPR g <-> row g + 8*(L/16), col L%16 per 16x16 tile
#pragma unroll
    for (int r = 0; r < 4; ++r) {
#pragma unroll
        for (int cc = 0; cc < 2; ++cc) {
            const int col = colBlock + wCol + cc * 16 + lm;
#pragma unroll
            for (int g = 0; g < 8; ++g) {
                const int row = rowBlock + wRow + r * 16 + g + 8 * lh;
                D[(size_t)row * MAT_N + col] = acc[r][cc][g];
            }
        }
    }
}

extern "C" void kernel_launch(void* const* d_in, const int* in_sizes, int n_in,
                              void* d_out, int out_size, void* d_ws, size_t ws_size,
                              hipStream_t stream) {
    (void)in_sizes; (void)n_in; (void)out_size; (void)ws_size;
    const float* x = (const float*)d_in[0];   // 4096x4096 fp32

    float* A = (float*)d_ws;                  // 64 MB basis
    float* T = A + (size_t)MAT_N * MAT_N;     // 64 MB intermediate (T[v][m])

    dct_basis_kernel<<<(MAT_N * MAT_N) / 256, 256, 0, stream>>>(A);

    dim3 grid(MAT_N / BLK, MAT_N / BLK);
    gemm_nt_wmma<<<grid, 256, 0, stream>>>(A, x, T);              // T = A * X^T
    gemm_nt_wmma<<<grid, 256, 0, stream>>>(A, T, (float*)d_out);  // C = A * T^T
}